// IFNode_35983236006443
// MI455X (gfx1250) — compile-verified
//
#include <hip/hip_runtime.h>
#include <stdint.h>

// ---------------------------------------------------------------------------
// Integrate-and-fire scan, X:[B=32,T=32,N=65536] f32 -> spikes same shape.
// Pure streaming workload (512 MB @ 23.3 TB/s ~= 22 us floor). Strategy:
//   - per-lane float4 (b128) fully-coalesced access along n
//   - T processed in 4 chunks of 8 rows, double-buffered in LDS via
//     CDNA5 async global->LDS DMA (ASYNCcnt), s_wait_asynccnt for sync
//   - non-temporal store hints for the write-once output stream
// ---------------------------------------------------------------------------

typedef float v4f __attribute__((ext_vector_type(4)));

#define BLOCK_THREADS 256
#define VEC           4
#define TILE_N        (BLOCK_THREADS * VEC)   // 1024 floats of n per block
#define T_STEPS       32
#define CHUNK         8
#define NCHUNKS       (T_STEPS / CHUNK)
#define LDS_V4F       (2 * CHUNK * BLOCK_THREADS)              // double buffer
#define LDS_BYTES     (LDS_V4F * (int)sizeof(v4f))             // 64 KB

// CDNA5 async DMA: global memory -> LDS, 16B per lane, tracked by ASYNCcnt.
__device__ __forceinline__ void async_load_b128(uint32_t lds_byte_addr,
                                                const float* gaddr) {
    asm volatile("global_load_async_to_lds_b128 %0, %1, off th:TH_LOAD_NT"
                 :
                 : "v"(lds_byte_addr), "v"((uint64_t)(uintptr_t)gaddr)
                 : "memory");
}

__device__ __forceinline__ void s_wait_asynccnt_8() {
    asm volatile("s_wait_asynccnt 8" ::: "memory");
}
__device__ __forceinline__ void s_wait_asynccnt_0() {
    asm volatile("s_wait_asynccnt 0" ::: "memory");
}

__global__ __launch_bounds__(BLOCK_THREADS)
void if_scan_kernel(const float* __restrict__ X, float* __restrict__ out) {
    constexpr int N = 65536;
    constexpr int TILES_PER_B = N / TILE_N;   // 64

    extern __shared__ v4f smem[];             // [2][CHUNK][BLOCK_THREADS] v4f

    const int tid  = threadIdx.x;
    const int b    = blockIdx.x / TILES_PER_B;
    const int tile = blockIdx.x % TILES_PER_B;
    const int n0   = tile * TILE_N + tid * VEC;

    const float* __restrict__ gin  = X   + (size_t)b * T_STEPS * N + n0;
    float*       __restrict__ gout = out + (size_t)b * T_STEPS * N + n0;

    // Raw LDS byte address (low 32 bits of the shared-aperture generic addr)
    const uint32_t ldsBase = (uint32_t)(uintptr_t)(&smem[0]);

    // Prologue: DMA chunk 0 (t = 0..7) into buffer 0.
#pragma unroll
    for (int r = 0; r < CHUNK; ++r) {
        async_load_b128(ldsBase + (uint32_t)((r * BLOCK_THREADS + tid) * sizeof(v4f)),
                        gin + (size_t)r * N);
    }

    float m0 = 0.0f, m1 = 0.0f, m2 = 0.0f, m3 = 0.0f;   // membrane potential

#pragma unroll
    for (int c = 0; c < NCHUNKS; ++c) {
        if (c + 1 < NCHUNKS) {
            // DMA next chunk into the other buffer, then wait for the
            // previous 8 (async loads complete in order -> cnt<=8 means
            // chunk c is fully resident in LDS).
            const int nb = (c + 1) & 1;
#pragma unroll
            for (int r = 0; r < CHUNK; ++r) {
                async_load_b128(
                    ldsBase + (uint32_t)(((nb * CHUNK + r) * BLOCK_THREADS + tid) * sizeof(v4f)),
                    gin + (size_t)((c + 1) * CHUNK + r) * N);
            }
            s_wait_asynccnt_8();
        } else {
            s_wait_asynccnt_0();
        }

        const int cb = c & 1;
#pragma unroll
        for (int r = 0; r < CHUNK; ++r) {
            const int t = c * CHUNK + r;
            v4f x = smem[(cb * CHUNK + r) * BLOCK_THREADS + tid]; // ds_load_b128

            m0 += x.x; m1 += x.y; m2 += x.z; m3 += x.w;

            const bool f0 = (m0 >= 1.0f);
            const bool f1 = (m1 >= 1.0f);
            const bool f2 = (m2 >= 1.0f);
            const bool f3 = (m3 >= 1.0f);

            v4f s;
            s.x = f0 ? 1.0f : 0.0f;
            s.y = f1 ? 1.0f : 0.0f;
            s.z = f2 ? 1.0f : 0.0f;
            s.w = f3 ? 1.0f : 0.0f;

            m0 = f0 ? 0.0f : m0;   // hard reset to V_RESET = 0
            m1 = f1 ? 0.0f : m1;
            m2 = f2 ? 0.0f : m2;
            m3 = f3 ? 0.0f : m3;

            // Write-once output stream: NT store (b128).
            __builtin_nontemporal_store(s, (v4f*)(gout + (size_t)t * N));
        }
    }
}

extern "C" void kernel_launch(void* const* d_in, const int* in_sizes, int n_in,
                              void* d_out, int out_size, void* d_ws, size_t ws_size,
                              hipStream_t stream) {
    (void)in_sizes; (void)n_in; (void)out_size; (void)d_ws; (void)ws_size;
    const float* X   = (const float*)d_in[0];
    float*       out = (float*)d_out;

    constexpr int B = 32;
    constexpr int N = 65536;
    const int grid = B * (N / TILE_N);   // 32 * 64 = 2048 blocks

    if_scan_kernel<<<grid, BLOCK_THREADS, LDS_BYTES, stream>>>(X, out);
}